// SingleStageDetector_2834678415613
// MI455X (gfx1250) — compile-verified
//
#include <hip/hip_runtime.h>
#include <hip/hip_bf16.h>

typedef __attribute__((ext_vector_type(2))) float v2f;
typedef __attribute__((ext_vector_type(8))) float v8f;

#define A_NUM 9
#define C_NUM 20
#define BATCH 32
#define CIN   256
#define HID   256
#define HGT   56
#define WID   56
#define HW    (HGT * WID)       // 3136
#define PTOT  (BATCH * HW)      // 100352
#define OCP   80                // padded out channels (65 -> 80)
#define NANCH (A_NUM * HW)      // 28224
#define MPOS  2048
#define NBOX  64

__device__ __forceinline__ v8f wmma_f32(v2f a, v2f b, v8f c) {
    // D = A(16x4, f32) * B(4x16, f32) + C(16x16, f32)
    return __builtin_amdgcn_wmma_f32_16x16x4_f32(false, a, false, b, (short)0, c,
                                                 false, false);
}

// ---------------------------------------------------------------------------
// Pad w2 (65x256) -> w2p (80x256, zero-filled) and b2 -> b2p so that WMMA
// N-tiles never need per-lane predication (EXEC must stay all-1s).
// ---------------------------------------------------------------------------
__global__ void prep_kernel(const float* __restrict__ w2, const float* __restrict__ b2,
                            float* __restrict__ w2p, float* __restrict__ b2p) {
    const int idx = blockIdx.x * blockDim.x + threadIdx.x;
    if (idx < OCP * HID) {
        const int o = idx / HID;
        const int c = idx - o * HID;
        w2p[idx] = (o < 65) ? w2[o * HID + c] : 0.0f;
    }
    if (idx < OCP) b2p[idx] = (idx < 65) ? b2[idx] : 0.0f;
}

// ---------------------------------------------------------------------------
// Fused conv1(1x1) -> leaky relu -> conv2(1x1), all via V_WMMA_F32_16X16X4_F32.
// 128 threads = 4 waves; each wave owns a 16-pixel M-tile. Activations are
// staged per-wave in LDS (wave-private slice -> no barriers needed).
// ---------------------------------------------------------------------------
__global__ __launch_bounds__(128) void gemm_fused_kernel(
    const float* __restrict__ feat, const float* __restrict__ w1,
    const float* __restrict__ b1, const float* __restrict__ w2p,
    const float* __restrict__ b2p, float* __restrict__ out2) {
    __shared__ float act[4][16][HID];   // 64 KB

    const int tid  = threadIdx.x;
    const int wave = tid >> 5;
    const int lane = tid & 31;
    const int lm   = lane & 15;   // M index (A) or N index (B/C)
    const int kh   = lane >> 4;   // selects K pair {0,1} vs {2,3}
    const int p0   = blockIdx.x * 64 + wave * 16;
    const int pl   = p0 + lm;     // HW % 16 == 0 -> tile never straddles batch
    const int bb   = pl / HW;
    const int hw   = pl - bb * HW;
    const size_t fbase = (size_t)bb * CIN * HW + hw;

    // -------- GEMM1: act = leakyrelu(feat @ w1^T + b1) --------
    for (int g = 0; g < 4; ++g) {
        v8f acc[4] = {};
        const int obase = g * 64;
        for (int k0 = 0; k0 < CIN; k0 += 4) {
            const int c0 = k0 + 2 * kh;
            v2f a;
            a.x = feat[fbase + (size_t)c0 * HW];
            a.y = feat[fbase + (size_t)(c0 + 1) * HW];
#pragma unroll
            for (int t = 0; t < 4; ++t) {
                const v2f bv = *(const v2f*)(w1 + (obase + t * 16 + lm) * CIN + c0);
                acc[t] = wmma_f32(a, bv, acc[t]);
            }
        }
#pragma unroll
        for (int t = 0; t < 4; ++t) {
            const int o = obase + t * 16 + lm;
            const float bias = b1[o];
#pragma unroll
            for (int r = 0; r < 8; ++r) {
                const int m = r + 8 * kh;           // C layout: M = r + 8*(lane/16)
                const float v = acc[t][r] + bias;
                act[wave][m][o] = (v > 0.0f) ? v : 0.01f * v;
            }
        }
    }

    // -------- GEMM2: out2 = act @ w2p^T + b2p --------
    v8f acc2[5] = {};
    for (int k0 = 0; k0 < HID; k0 += 4) {
        const int c0 = k0 + 2 * kh;
        const v2f a = *(const v2f*)(&act[wave][lm][c0]);
#pragma unroll
        for (int t = 0; t < 5; ++t) {
            const v2f bv = *(const v2f*)(w2p + (t * 16 + lm) * HID + c0);
            acc2[t] = wmma_f32(a, bv, acc2[t]);
        }
    }
#pragma unroll
    for (int t = 0; t < 5; ++t) {
        const int o = t * 16 + lm;
        const float bias = b2p[o];
#pragma unroll
        for (int r = 0; r < 8; ++r) {
            const int p = p0 + r + 8 * kh;
            out2[(size_t)p * OCP + o] = acc2[t][r] + bias;
        }
    }
}

// ---------------------------------------------------------------------------
// IoU: block = 4 proposals x 64 GT boxes. Coalesced stores (m contiguous).
// ---------------------------------------------------------------------------
__global__ __launch_bounds__(256) void iou_kernel(
    const float* __restrict__ out2, const float* __restrict__ gridp,
    const float* __restrict__ anc, const float* __restrict__ bboxes,
    float* __restrict__ iou_out) {
    __shared__ float sb[NBOX][4];
    __shared__ float sgarea[NBOX];
    __shared__ float sprop[4][5];

    const int tid = threadIdx.x;
    const int b   = blockIdx.y;
    const int n0  = blockIdx.x * 4;

    if (tid < NBOX) {
        const float* g = bboxes + ((size_t)b * NBOX + tid) * 5;
        const float x1 = g[0], y1 = g[1], x2 = g[2], y2 = g[3];
        sb[tid][0] = x1; sb[tid][1] = y1; sb[tid][2] = x2; sb[tid][3] = y2;
        sgarea[tid] = (x2 - x1) * (y2 - y1);
    }
    if (tid < 4) {
        const int n  = n0 + tid;
        const int a  = n / HW;
        const int hw = n - a * HW;
        const int h  = hw / WID;
        const int w  = hw - h * WID;
        const float* o = out2 + ((size_t)b * HW + hw) * OCP + a * 5;
        const float tx = o[1], ty = o[2], tw = o[3], th = o[4];
        const size_t gi = (((size_t)b * HGT + h) * WID + w) * 2;
        const float gx = gridp[gi], gy = gridp[gi + 1];
        const float aw = anc[a * 2], ah = anc[a * 2 + 1];
        const float nw = aw * __expf(tw);
        const float nh = ah * __expf(th);
        const float cx = gx + tx, cy = gy + ty;
        const float px1 = cx - 0.5f * nw, py1 = cy - 0.5f * nh;
        const float px2 = cx + 0.5f * nw, py2 = cy + 0.5f * nh;
        sprop[tid][0] = px1; sprop[tid][1] = py1;
        sprop[tid][2] = px2; sprop[tid][3] = py2;
        sprop[tid][4] = (px2 - px1) * (py2 - py1);
    }
    __syncthreads();

    const int ln = tid >> 6;   // 0..3
    const int m  = tid & 63;
    const float px1 = sprop[ln][0], py1 = sprop[ln][1];
    const float px2 = sprop[ln][2], py2 = sprop[ln][3], pa = sprop[ln][4];
    const float ix1 = fmaxf(px1, sb[m][0]);
    const float iy1 = fmaxf(py1, sb[m][1]);
    const float ix2 = fminf(px2, sb[m][2]);
    const float iy2 = fminf(py2, sb[m][3]);
    const float inter = fmaxf(ix2 - ix1, 0.0f) * fmaxf(iy2 - iy1, 0.0f);
    const float iou = inter / (sgarea[m] + pa - inter);
    iou_out[(((size_t)b * NANCH) + (n0 + ln)) * NBOX + m] = iou;
}

// ---------------------------------------------------------------------------
// class_pos gather: class_pos[i][t] = x2[bi, 45+t, hi, wi]
// ---------------------------------------------------------------------------
__global__ void classpos_kernel(const float* __restrict__ out2,
                                const int* __restrict__ pos,
                                float* __restrict__ cls_out) {
    const int i = blockIdx.x * blockDim.x + threadIdx.x;
    if (i >= MPOS) return;
    const int idx = pos[i];
    const int bi  = idx / NANCH;
    const int hw  = idx % HW;
    const float* src = out2 + ((size_t)bi * HW + hw) * OCP + 45;
#pragma unroll
    for (int t = 0; t < C_NUM; ++t) cls_out[(size_t)i * C_NUM + t] = src[t];
}

// ---------------------------------------------------------------------------
// conf loss + reg loss, single-block reduction
// ---------------------------------------------------------------------------
__global__ __launch_bounds__(256) void loss_kernel(
    const float* __restrict__ out2, const float* __restrict__ GT,
    const int* __restrict__ pos, const int* __restrict__ neg,
    float* __restrict__ loss_out) {
    __shared__ float sc[256];
    __shared__ float sr[256];
    const int tid = threadIdx.x;
    float csum = 0.0f, rsum = 0.0f;
    for (int i = tid; i < MPOS; i += 256) {
        {
            const int idx = pos[i];
            const int b   = idx / NANCH;
            const int rem = idx - b * NANCH;
            const int a   = rem / HW;
            const int hw  = rem - a * HW;
            const float* src = out2 + ((size_t)b * HW + hw) * OCP + a * 5;
            const float conf = 1.0f / (1.0f + __expf(-src[0]));
            const float dc = conf - 1.0f;
            csum += dc * dc;
#pragma unroll
            for (int k = 0; k < 4; ++k) {
                const float d = src[1 + k] - GT[(size_t)i * 4 + k];
                rsum += d * d;
            }
        }
        {
            const int idx = neg[i];
            const int b   = idx / NANCH;
            const int rem = idx - b * NANCH;
            const int a   = rem / HW;
            const int hw  = rem - a * HW;
            const float conf =
                1.0f / (1.0f + __expf(-out2[((size_t)b * HW + hw) * OCP + a * 5]));
            csum += conf * conf;
        }
    }
    sc[tid] = csum;
    sr[tid] = rsum;
    __syncthreads();
    for (int s = 128; s > 0; s >>= 1) {
        if (tid < s) { sc[tid] += sc[tid + s]; sr[tid] += sr[tid + s]; }
        __syncthreads();
    }
    if (tid == 0) loss_out[0] = sc[0] / (2.0f * MPOS) + sr[0] / MPOS;
}

// ---------------------------------------------------------------------------
extern "C" void kernel_launch(void* const* d_in, const int* in_sizes, int n_in,
                              void* d_out, int out_size, void* d_ws, size_t ws_size,
                              hipStream_t stream) {
    (void)in_sizes; (void)n_in; (void)out_size; (void)ws_size;
    const float* feat = (const float*)d_in[0];
    const float* w1   = (const float*)d_in[1];
    const float* b1   = (const float*)d_in[2];
    const float* w2   = (const float*)d_in[3];
    const float* b2   = (const float*)d_in[4];
    const float* anc  = (const float*)d_in[5];
    const float* grd  = (const float*)d_in[6];
    const float* bbx  = (const float*)d_in[7];
    const float* GT   = (const float*)d_in[8];
    const int*   pos  = (const int*)d_in[9];
    const int*   neg  = (const int*)d_in[10];
    float* out = (float*)d_out;

    float* out2 = (float*)d_ws;                       // PTOT*OCP fp32 (32.1 MB)
    float* w2p  = out2 + (size_t)PTOT * OCP;          // OCP*HID
    float* b2p  = w2p + (size_t)OCP * HID;            // OCP

    prep_kernel<<<(OCP * HID + 255) / 256, 256, 0, stream>>>(w2, b2, w2p, b2p);
    gemm_fused_kernel<<<PTOT / 64, 128, 0, stream>>>(feat, w1, b1, w2p, b2p, out2);
    iou_kernel<<<dim3(NANCH / 4, BATCH), 256, 0, stream>>>(out2, grd, anc, bbx, out + 1);
    classpos_kernel<<<(MPOS + 255) / 256, 256, 0, stream>>>(
        out2, pos, out + 1 + (size_t)BATCH * NANCH * NBOX);
    loss_kernel<<<1, 256, 0, stream>>>(out2, GT, pos, neg, out);
}